// GlobalChannelCovariancePooling_25031069401649
// MI455X (gfx1250) — compile-verified
//
#include <hip/hip_runtime.h>

typedef float v2f __attribute__((ext_vector_type(2)));
typedef float v8f __attribute__((ext_vector_type(8)));

#define B_DIM  32
#define C_DIM  256
#define N_DIM  4096
#define TM     128         // macro-tile rows (M channels)
#define TN     64          // macro-tile cols (N channels)
#define KC     64          // K chunk staged in LDS
#define LDA    68          // padded LDS row stride (floats): 16B-aligned rows, conflict-free b64 reads

__global__ __launch_bounds__(256)
void cov_wmma_f32_kernel(const float* __restrict__ x, float* __restrict__ out) {
    __shared__ float ldsA[2][TM * LDA];    // double-buffered 128 x 68
    __shared__ float ldsB[2][TN * LDA];    // double-buffered 64 x 68
    __shared__ float partA[TM * 2];
    __shared__ float partB[TN * 4];
    __shared__ float sumsA[TM];
    __shared__ float sumsB[TN];

    const int b    = blockIdx.x >> 3;          // batch
    const int tile = blockIdx.x & 7;           // 2 (M) x 4 (N) macro-tiles
    const int row0 = (tile >> 2) * TM;         // M channel base
    const int col0 = (tile & 3)  * TN;         // N channel base

    const float* Xb    = x + (size_t)b * C_DIM * N_DIM;
    const float* Arows = Xb + (size_t)row0 * N_DIM;
    const float* Brows = Xb + (size_t)col0 * N_DIM;

    const int t = threadIdx.x;                 // 0..255

    // staging assignments (async copy + sum readback)
    const int arow  = t >> 1;                  // 0..127, A panel row
    const int ahalf = t & 1;                   // 0..1   -> cols [ahalf*32, +32) : 8 x b128
    const int brow  = t >> 2;                  // 0..63,  B panel row
    const int bquad = t & 3;                   // 0..3   -> cols [bquad*16, +16) : 4 x b128

    // wave/compute assignments: wave grid 4 (M) x 2 (N); each wave: 32x32 = 2x2 subtiles
    const int lane = t & 31;
    const int wave = t >> 5;                   // 0..7
    const int wm   = (wave & 3) * 32;          // wave M base within macro-tile
    const int wn   = (wave >> 2) * 32;         // wave N base within macro-tile

    const int lidx  = lane & 15;
    const int khalf = (lane >> 4) * 2;         // K-pair base per half-wave

    const int a_off0 = (wm + lidx) * LDA + khalf;        // LDS operand offsets (floats)
    const int a_off1 = (wm + 16 + lidx) * LDA + khalf;
    const int b_off0 = (wn + lidx) * LDA + khalf;
    const int b_off1 = (wn + 16 + lidx) * LDA + khalf;

    const int a_stage = arow * LDA + ahalf * 32;         // staging offsets (floats)
    const int b_stage = brow * LDA + bquad * 16;
    const unsigned alds0 = (unsigned)(uintptr_t)&ldsA[0][a_stage];
    const unsigned blds0 = (unsigned)(uintptr_t)&ldsB[0][b_stage];
    const unsigned abufstep = (unsigned)(TM * LDA * sizeof(float));
    const unsigned bbufstep = (unsigned)(TN * LDA * sizeof(float));

    v8f acc00 = {}, acc01 = {}, acc10 = {}, acc11 = {};
    float sA = 0.0f, sB = 0.0f;

    // ---- async stage: global memory into the LDS buffer chosen by bufsel, 16B per lane per op (ASYNCcnt) ----
    // INST_OFFSET is added to both the global and LDS addresses (ISA 15.18.3),
    // so one base address pair serves all ops of a panel.
    auto issue_chunk = [&](int k0, int bufsel) {
        const float* agp = Arows + (size_t)arow * N_DIM + k0 + ahalf * 32;
        const float* bgp = Brows + (size_t)brow * N_DIM + k0 + bquad * 16;
        const unsigned al = alds0 + (unsigned)bufsel * abufstep;
        const unsigned bl = blds0 + (unsigned)bufsel * bbufstep;
#pragma unroll
        for (int i = 0; i < 8; ++i) {
            asm volatile("global_load_async_to_lds_b128 %0, %1, off offset:%c2"
                         :: "v"(al), "v"(agp), "i"(16 * i) : "memory");
        }
#pragma unroll
        for (int i = 0; i < 4; ++i) {
            asm volatile("global_load_async_to_lds_b128 %0, %1, off offset:%c2"
                         :: "v"(bl), "v"(bgp), "i"(16 * i) : "memory");
        }
    };

    issue_chunk(0, 0);                         // prologue: fill buffer 0

    int buf = 0;
    for (int k0 = 0; k0 < N_DIM; k0 += KC, buf ^= 1) {
        asm volatile("s_wait_asynccnt 0x0" ::: "memory");   // chunk k landed (my waves' ops)
        __syncthreads();   // all waves landed; all reads of buf^1 finished last iter

        if (k0 + KC < N_DIM) {
            issue_chunk(k0 + KC, buf ^ 1);     // DMA next chunk during this chunk's WMMAs
            if (k0 + 2 * KC < N_DIM) {         // warm GL2 two chunks ahead
                __builtin_prefetch(Arows + (size_t)arow * N_DIM + k0 + 2 * KC + ahalf * 32, 0, 0);
                __builtin_prefetch(Brows + (size_t)brow * N_DIM + k0 + 2 * KC + bquad * 16, 0, 0);
            }
        }

        const float* La = ldsA[buf];
        const float* Lb = ldsB[buf];

        // channel sums: read back the tile this thread just DMA'd
#pragma unroll
        for (int i = 0; i < 8; ++i) {
            float4 v = *(const float4*)(La + a_stage + 4 * i);
            sA += v.x + v.y + v.z + v.w;
        }
#pragma unroll
        for (int i = 0; i < 4; ++i) {
            float4 v = *(const float4*)(Lb + b_stage + 4 * i);
            sB += v.x + v.y + v.z + v.w;
        }

#pragma unroll
        for (int kk = 0; kk < KC; kk += 4) {
            v2f av0 = *(const v2f*)(La + a_off0 + kk);
            v2f av1 = *(const v2f*)(La + a_off1 + kk);
            v2f bv0 = *(const v2f*)(Lb + b_off0 + kk);
            v2f bv1 = *(const v2f*)(Lb + b_off1 + kk);
            acc00 = __builtin_amdgcn_wmma_f32_16x16x4_f32(
                        false, av0, false, bv0, (short)0, acc00, false, false);
            acc01 = __builtin_amdgcn_wmma_f32_16x16x4_f32(
                        false, av0, false, bv1, (short)0, acc01, false, false);
            acc10 = __builtin_amdgcn_wmma_f32_16x16x4_f32(
                        false, av1, false, bv0, (short)0, acc10, false, false);
            acc11 = __builtin_amdgcn_wmma_f32_16x16x4_f32(
                        false, av1, false, bv1, (short)0, acc11, false, false);
        }
    }

    // Reduce per-thread partial row sums -> per-channel sums for this tile
    partA[arow * 2 + ahalf] = sA;
    partB[brow * 4 + bquad] = sB;
    __syncthreads();
    if (t < TM) {
        sumsA[t] = partA[t * 2] + partA[t * 2 + 1];
    } else if (t < TM + TN) {
        const int r = t - TM;
        sumsB[r] = partB[r * 4] + partB[r * 4 + 1] + partB[r * 4 + 2] + partB[r * 4 + 3];
    }
    __syncthreads();

    // cov[m,n] = (S[m,n] - sumA[m]*sumB[n]/N) / (N-1)
    const float invN   = 1.0f / (float)N_DIM;
    const float invNm1 = 1.0f / (float)(N_DIM - 1);
    float* outb = out + (size_t)b * C_DIM * C_DIM;

    const int mb0 = wm + ((lane >> 4) << 3);        // D layout: lanes 16-31 hold M+8
    const int mb1 = mb0 + 16;
    const int nl0 = wn + lidx;
    const int nl1 = nl0 + 16;
    const float sBn0 = sumsB[nl0] * invN;
    const float sBn1 = sumsB[nl1] * invN;

#pragma unroll
    for (int r = 0; r < 8; ++r) {
        const int   ml0  = mb0 + r;
        const int   ml1  = mb1 + r;
        const float sA0  = sumsA[ml0];
        const float sA1  = sumsA[ml1];
        float* o0 = outb + (size_t)(row0 + ml0) * C_DIM + col0;
        float* o1 = outb + (size_t)(row0 + ml1) * C_DIM + col0;
        o0[nl0] = (acc00[r] - sA0 * sBn0) * invNm1;
        o0[nl1] = (acc01[r] - sA0 * sBn1) * invNm1;
        o1[nl0] = (acc10[r] - sA1 * sBn0) * invNm1;
        o1[nl1] = (acc11[r] - sA1 * sBn1) * invNm1;
    }
}

extern "C" void kernel_launch(void* const* d_in, const int* in_sizes, int n_in,
                              void* d_out, int out_size, void* d_ws, size_t ws_size,
                              hipStream_t stream) {
    (void)in_sizes; (void)n_in; (void)out_size; (void)d_ws; (void)ws_size;
    const float* x = (const float*)d_in[0];
    float* out = (float*)d_out;
    dim3 grid(B_DIM * (C_DIM / TM) * (C_DIM / TN));   // 32 * 2 * 4 = 256
    dim3 block(256);
    hipLaunchKernelGGL(cov_wmma_f32_kernel, grid, block, 0, stream, x, out);
}